// Quantize_31155692765408
// MI455X (gfx1250) — compile-verified
//
#include <hip/hip_runtime.h>
#include <hip/hip_bf16.h>

typedef __attribute__((ext_vector_type(2))) float v2f;
typedef __attribute__((ext_vector_type(8))) float v8f;

#define CB_SIZE   1024          // codebook entries
#define CODE_DIM  256           // feature dim
#define HW        1024          // 32*32 spatial
#define NQ        16777216      // 64*256*1024 elements per output tensor

// ---------------------------------------------------------------------------
// Kernel 1: ||e||^2 per codebook entry -> d_ws (1024 floats)
// ---------------------------------------------------------------------------
__global__ __launch_bounds__(256)
void vq_enorm_kernel(const float* __restrict__ cb, float* __restrict__ en) {
    int e = blockIdx.x * blockDim.x + threadIdx.x;   // 0..1023
    const float4* p = (const float4*)(cb + (size_t)e * CODE_DIM);
    float s = 0.0f;
    #pragma unroll 8
    for (int i = 0; i < CODE_DIM / 4; ++i) {
        float4 q = p[i];
        s = fmaf(q.x, q.x, s);
        s = fmaf(q.y, q.y, s);
        s = fmaf(q.z, q.z, s);
        s = fmaf(q.w, q.w, s);
    }
    en[e] = s;
}

// ---------------------------------------------------------------------------
// Kernel 2: distances via V_WMMA_F32_16X16X4_F32, argmin, gather, write.
// Grid: 1024 blocks x 128 threads (4 waves). Block owns 64 consecutive rows.
// Wave w owns codebook columns [w*256, w*256+256)  (disjoint -> 1MB cb/WG).
// ---------------------------------------------------------------------------
__global__ __launch_bounds__(128)
void vq_main_kernel(const float* __restrict__ z,
                    const float* __restrict__ cb,
                    const float* __restrict__ en,
                    float* __restrict__ out) {
    __shared__ float red_v[4][64];
    __shared__ int   red_i[4][64];
    __shared__ int   fidx[64];

    const int tid  = threadIdx.x;
    const int lane = tid & 31;          // wave32
    const int wave = tid >> 5;
    const int m    = lane & 15;         // M (A) / N (B) / col within tile (C)
    const int kh   = lane >> 4;         // K half-select

    const int    n0    = blockIdx.x * 64;          // first flat row
    const int    b     = n0 >> 10;                 // batch image
    const int    s0    = n0 & 1023;                // spatial offset
    const size_t zbase = (size_t)b * (CODE_DIM * HW) + (size_t)s0;

    // Preload all 16 ||e||^2 values this lane will need (one clause,
    // avoids a dependent load + full loadcnt drain at every tile epilogue).
    float e2arr[16];
    {
        const float* enp = en + wave * 256 + m;
        #pragma unroll
        for (int ct = 0; ct < 16; ++ct) e2arr[ct] = enp[ct * 16];
    }

    float minv[4][8];
    int   mini[4][8];
    #pragma unroll
    for (int r = 0; r < 4; ++r)
        #pragma unroll
        for (int j = 0; j < 8; ++j) { minv[r][j] = 3.4e38f; mini[r][j] = 0; }

    // A-fragment base: lane reads z_flat[row = rtile*16 + m][k0 + 2*kh (+1)]
    // z_flat[n][k] = z[zbase + k*HW + (n - n0)]
    const float* zp = z + zbase + (size_t)(2 * kh) * HW + m;

    for (int ct = 0; ct < 16; ++ct) {
        const int colbase = wave * 256 + ct * 16;
        // B-fragment: lane reads codebook[colbase + m][k0 + 2*kh .. +1] (contig pair)
        const float* cbp = cb + (size_t)(colbase + m) * CODE_DIM + 2 * kh;

        v8f acc[4] = {};
        #pragma unroll 4
        for (int k0 = 0; k0 < CODE_DIM; k0 += 4) {
            v2f bf = *(const v2f*)(cbp + k0);
            const float* zk = zp + (size_t)k0 * HW;
            #pragma unroll
            for (int r = 0; r < 4; ++r) {
                v2f a;
                a.x = zk[r * 16];
                a.y = zk[r * 16 + HW];
                acc[r] = __builtin_amdgcn_wmma_f32_16x16x4_f32(
                    false, a, false, bf, (short)0, acc[r], false, false);
            }
        }

        const float e2  = e2arr[ct];
        const int   col = colbase + m;
        #pragma unroll
        for (int r = 0; r < 4; ++r)
            #pragma unroll
            for (int j = 0; j < 8; ++j) {
                float d = fmaf(-2.0f, acc[r][j], e2);   // ||e||^2 - 2 z.e
                if (d < minv[r][j]) { minv[r][j] = d; mini[r][j] = col; }
            }
    }

    // Butterfly argmin across each 16-lane half (cols of the tile),
    // lower index wins ties (matches jnp.argmin).
    #pragma unroll
    for (int r = 0; r < 4; ++r)
        #pragma unroll
        for (int j = 0; j < 8; ++j) {
            float v = minv[r][j];
            int   i = mini[r][j];
            #pragma unroll
            for (int mask = 8; mask >= 1; mask >>= 1) {
                float ov = __shfl_xor(v, mask, 32);
                int   oi = __shfl_xor(i, mask, 32);
                if (ov < v || (ov == v && oi < i)) { v = ov; i = oi; }
            }
            if (m == 0) {
                int row64 = r * 16 + j + 8 * kh;   // C-layout: row = j + 8*(lane>=16)
                red_v[wave][row64] = v;
                red_i[wave][row64] = i;
            }
        }
    __syncthreads();

    // Cross-wave reduction (waves own ascending disjoint column ranges).
    if (tid < 64) {
        float v = red_v[0][tid];
        int   i = red_i[0][tid];
        #pragma unroll
        for (int w = 1; w < 4; ++w) {
            float ov = red_v[w][tid];
            int   oi = red_i[w][tid];
            if (ov < v || (ov == v && oi < i)) { v = ov; i = oi; }
        }
        fidx[tid] = i;
        // indices as f32; non-temporal: never re-read by us, keep L2 for cb/z
        __builtin_nontemporal_store((float)i, &out[(size_t)2 * NQ + n0 + tid]);
    }
    __syncthreads();

    // Gather + NCHW write: for each channel c, rows 0..63 are contiguous.
    // quantized and straight_through are numerically identical in forward.
    #pragma unroll 1
    for (int c0 = 0; c0 < CODE_DIM; c0 += 2) {
        int   c   = c0 + (tid >> 6);
        int   row = tid & 63;
        float v   = cb[(size_t)fidx[row] * CODE_DIM + c];
        size_t off = (size_t)b * (CODE_DIM * HW) + (size_t)c * HW + s0 + row;
        __builtin_nontemporal_store(v, &out[off]);                 // quantized
        __builtin_nontemporal_store(v, &out[(size_t)NQ + off]);    // straight_through
    }
}

extern "C" void kernel_launch(void* const* d_in, const int* in_sizes, int n_in,
                              void* d_out, int out_size, void* d_ws, size_t ws_size,
                              hipStream_t stream) {
    const float* z  = (const float*)d_in[0];   // [64,256,32,32]
    const float* cb = (const float*)d_in[1];   // [1024,256]
    float* en  = (float*)d_ws;                 // 1024 floats scratch
    float* out = (float*)d_out;                // [NQ] + [NQ] + [65536]

    vq_enorm_kernel<<<CB_SIZE / 256, 256, 0, stream>>>(cb, en);
    vq_main_kernel<<<65536 / 64, 128, 0, stream>>>(z, cb, en, out);
}